// RNNDecoder_51462298141083
// MI455X (gfx1250) — compile-verified
//
#include <hip/hip_runtime.h>
#include <math.h>

// Problem constants (fixed by the reference).
#define B_ 16
#define S_ 256
#define H_ 512
#define T_ 24
// gates: 4*H = 2048 columns; W_ih has IN = H+1 = 513 columns.

typedef float v2f __attribute__((ext_vector_type(2)));
typedef float v8f __attribute__((ext_vector_type(8)));

__device__ __forceinline__ float sigmoidf_(float x) { return 1.0f / (1.0f + expf(-x)); }

// ---------------------------------------------------------------------------
// fp32 WMMA tile accumulation: acc(16x16) += A(16xK) * W(16xK)^T
// V_WMMA_F32_16X16X4_F32 operand layout (ISA 7.12.2, 32-bit A 16x4):
//   lane l: row r = l&15 ; K sub-offset = 2*(l>>4) ; vgpr j adds +j to K.
// B (4x16) mirrors this with N = l&15. Both operands load the contiguous
// pair [row*ld + k0 + 2*(lane>>4) + {0,1}]  (merges into global_load_b64).
// C/D: vgpr i holds element (m = i + 8*(lane>>4), n = lane&15).
// ---------------------------------------------------------------------------
__device__ __forceinline__ void wmma_acc_f32(const float* __restrict__ aPtr,
                                             const float* __restrict__ wPtr,
                                             int K, v8f& acc) {
  for (int k0 = 0; k0 < K; k0 += 4) {
    v2f a, b;
    a.x = aPtr[k0];
    a.y = aPtr[k0 + 1];
    b.x = wPtr[k0];
    b.y = wPtr[k0 + 1];
    acc = __builtin_amdgcn_wmma_f32_16x16x4_f32(false, a, false, b, (short)0, acc,
                                                false, false);
  }
}

// ---------------------------------------------------------------------------
// Init: h = h0, c = c0, inp = 0   (fresh state every call: deterministic)
// ---------------------------------------------------------------------------
__global__ void k_init(const float* __restrict__ h0, const float* __restrict__ c0,
                       float* __restrict__ h, float* __restrict__ c,
                       float* __restrict__ inp) {
  int i = blockIdx.x * blockDim.x + threadIdx.x;
  if (i < B_ * H_) {
    h[i] = h0[i];
    c[i] = c0[i];
  }
  if (i < B_) inp[i] = 0.0f;
}

// ---------------------------------------------------------------------------
// enc_proj[(b,s),k] = sum_h enc[(b,s),h] * W_e[k,h]   (W_e = W_attn[:,512:])
// 4096x512 @ 512x512 -> 8192 tiles of 16x16, one wave per tile. Run once.
// ---------------------------------------------------------------------------
__global__ void k_enc_proj(const float* __restrict__ enc,
                           const float* __restrict__ W_attn,
                           float* __restrict__ enc_proj) {
  int wave = (blockIdx.x * blockDim.x + threadIdx.x) >> 5;
  int lane = threadIdx.x & 31;
  if (wave >= (B_ * S_ / 16) * (H_ / 16)) return;
  int mt = wave >> 5;  // 256 M-tiles
  int nt = wave & 31;  // 32 N-tiles
  int r = lane & 15;
  int koff = 2 * (lane >> 4);
  const float* aPtr = enc + (size_t)(mt * 16 + r) * H_ + koff;
  const float* wPtr = W_attn + (size_t)(nt * 16 + r) * (2 * H_) + H_ + koff;
  v8f acc = {};
  wmma_acc_f32(aPtr, wPtr, H_, acc);
  int n = lane & 15;
  int mb = 8 * (lane >> 4);
  for (int i = 0; i < 8; ++i)
    enc_proj[(size_t)(mt * 16 + mb + i) * H_ + nt * 16 + n] = acc[i];
}

// ---------------------------------------------------------------------------
// q[b,k] = sum_h h[b,h] * W_h[k,h] + b_attn[k]   (W_h = W_attn[:,:512])
// 32 tiles = 32 waves = 1 block of 1024 threads.
// ---------------------------------------------------------------------------
__global__ void k_query(const float* __restrict__ h, const float* __restrict__ W_attn,
                        const float* __restrict__ b_attn, float* __restrict__ q) {
  int nt = threadIdx.x >> 5;  // 0..31
  int lane = threadIdx.x & 31;
  int r = lane & 15;
  int koff = 2 * (lane >> 4);
  const float* aPtr = h + r * H_ + koff;
  const float* wPtr = W_attn + (size_t)(nt * 16 + r) * (2 * H_) + koff;
  v8f acc = {};
  wmma_acc_f32(aPtr, wPtr, H_, acc);
  int n = lane & 15;
  int mb = 8 * (lane >> 4);
  int ng = nt * 16 + n;
  float bias = b_attn[ng];
  for (int i = 0; i < 8; ++i) q[(mb + i) * H_ + ng] = acc[i] + bias;
}

// ---------------------------------------------------------------------------
// Fused attention: scores -> softmax -> ctx, one block per batch b.
// 512 threads = 16 waves. Phase 1: wave w computes score rows s=16w..16w+15
// (q[b,:], v staged in registers: each lane reuses its 16 strided values).
// Phase 2: block softmax over S=256. Phase 3: thread t = ctx column h.
// ---------------------------------------------------------------------------
__global__ void k_attn(const float* __restrict__ enc_proj, const float* __restrict__ q,
                       const float* __restrict__ v, const float* __restrict__ enc,
                       float* __restrict__ ctx, float* __restrict__ xcat) {
  __shared__ float sa[S_];
  __shared__ float red[S_];
  int b = blockIdx.x;
  int t = threadIdx.x;
  int wave = t >> 5;
  int lane = t & 31;

  // Stage q-row and v per lane (16 strided values each, reused for 16 rows).
  float qreg[16], vreg[16];
  for (int i = 0; i < 16; ++i) {
    qreg[i] = q[b * H_ + lane + 32 * i];
    vreg[i] = v[lane + 32 * i];
  }

  // Phase 1: scores[b,s] = sum_h relu(enc_proj[b,s,h] + q[b,h]) * v[h]
  for (int s0 = 0; s0 < 16; ++s0) {
    int s = wave * 16 + s0;
    const float* ep = enc_proj + ((size_t)b * S_ + s) * H_;
    __builtin_prefetch(ep + H_ + lane * 16, 0, 1);  // next row -> global_prefetch_b8
    float p = 0.0f;
    for (int i = 0; i < 16; ++i)
      p += fmaxf(ep[lane + 32 * i] + qreg[i], 0.0f) * vreg[i];
    for (int m = 16; m > 0; m >>= 1) p += __shfl_xor(p, m, 32);
    if (lane == 0) sa[s] = p;
  }
  __syncthreads();

  // Phase 2: softmax over S
  if (t < S_) red[t] = sa[t];
  __syncthreads();
  for (int off = S_ / 2; off > 0; off >>= 1) {
    if (t < off) red[t] = fmaxf(red[t], red[t + off]);
    __syncthreads();
  }
  float mx = red[0];
  __syncthreads();
  if (t < S_) {
    float e = expf(sa[t] - mx);
    sa[t] = e;
    red[t] = e;
  }
  __syncthreads();
  for (int off = S_ / 2; off > 0; off >>= 1) {
    if (t < off) red[t] += red[t + off];
    __syncthreads();
  }
  float inv = 1.0f / red[0];
  __syncthreads();
  if (t < S_) sa[t] *= inv;
  __syncthreads();

  // Phase 3: ctx[b,h] = sum_s a[s]*enc[b,s,h]; thread t owns column h=t.
  float acc = 0.0f;
  const float* eb = enc + (size_t)b * S_ * H_;
  for (int s = 0; s < S_; ++s) acc += sa[s] * eb[s * H_ + t];
  ctx[b * H_ + t] = acc;
  xcat[b * (2 * H_) + H_ + t] = acc;
}

// ---------------------------------------------------------------------------
// Fused gates GEMM + LSTM. One block (4 waves) per 16-column group g (32
// groups). Wave w computes the 16x16 tile of gate w (columns g*16..g*16+15):
//   gates = ctx@W_ih[:,1:]^T + h@W_hh^T + inp*W_ih[:,0] + b_ih + b_hh
// into LDS, then the block applies the LSTM elementwise for its columns.
// Reads h_cur, writes h_next (double-buffered: other blocks still read h_cur).
// ---------------------------------------------------------------------------
__global__ void k_gates_lstm(const float* __restrict__ ctx,
                             const float* __restrict__ h_cur,
                             const float* __restrict__ inp,
                             const float* __restrict__ W_ih,
                             const float* __restrict__ W_hh,
                             const float* __restrict__ b_ih,
                             const float* __restrict__ b_hh,
                             float* __restrict__ c, float* __restrict__ h_next,
                             float* __restrict__ xcat) {
  __shared__ float gl[4][16][16];
  int g = blockIdx.x;           // column group 0..31
  int wave = threadIdx.x >> 5;  // gate index 0..3 (i,f,g,o)
  int lane = threadIdx.x & 31;
  int nt = wave * 32 + g;       // tile over gates' 2048 columns
  int r = lane & 15;
  int koff = 2 * (lane >> 4);

  v8f acc = {};
  // ctx @ W_ih[:,1:]^T   (row stride 513)
  wmma_acc_f32(ctx + r * H_ + koff,
               W_ih + (size_t)(nt * 16 + r) * (H_ + 1) + 1 + koff, H_, acc);
  // h @ W_hh^T
  wmma_acc_f32(h_cur + r * H_ + koff,
               W_hh + (size_t)(nt * 16 + r) * H_ + koff, H_, acc);

  int n = lane & 15;
  int mb = 8 * (lane >> 4);
  int ng = nt * 16 + n;
  float w0 = W_ih[(size_t)ng * (H_ + 1)];  // column 0 multiplies scalar inp
  float bias = b_ih[ng] + b_hh[ng];
  for (int i = 0; i < 8; ++i) gl[wave][mb + i][n] = acc[i] + inp[mb + i] * w0 + bias;
  __syncthreads();

  // LSTM elementwise: 256 elements (16 batches x 16 columns), 128 threads.
  for (int e = threadIdx.x; e < 256; e += 128) {
    int m = e >> 4;
    int jj = e & 15;
    int j = g * 16 + jj;
    float ig = gl[0][m][jj];
    float fg = gl[1][m][jj];
    float gg = gl[2][m][jj];
    float og = gl[3][m][jj];
    float c2 = sigmoidf_(fg) * c[m * H_ + j] + sigmoidf_(ig) * tanhf(gg);
    float h2 = sigmoidf_(og) * tanhf(c2);
    c[m * H_ + j] = c2;
    h_next[m * H_ + j] = h2;
    xcat[m * (2 * H_) + j] = h2;
  }
}

// ---------------------------------------------------------------------------
// Fused head: mid = relu(xcat@W1^T + b1) (16 tiles, K=1024) into LDS,
// then out[b] = mid[b,:]@W2 + b2. One block, 512 threads = 16 waves.
// ---------------------------------------------------------------------------
__global__ void k_head(const float* __restrict__ xcat, const float* __restrict__ W1,
                       const float* __restrict__ b1, const float* __restrict__ W2,
                       const float* __restrict__ b2, float* __restrict__ out,
                       float* __restrict__ inp, int t) {
  __shared__ float ml[16][H_ / 2];
  int wave = threadIdx.x >> 5;  // tile / batch index
  int lane = threadIdx.x & 31;
  int r = lane & 15;
  int koff = 2 * (lane >> 4);

  v8f acc = {};
  wmma_acc_f32(xcat + r * (2 * H_) + koff,
               W1 + (size_t)(wave * 16 + r) * (2 * H_) + koff, 2 * H_, acc);
  int n = lane & 15;
  int mb = 8 * (lane >> 4);
  int ng = wave * 16 + n;
  float bias = b1[ng];
  for (int i = 0; i < 8; ++i) ml[mb + i][ng] = fmaxf(acc[i] + bias, 0.0f);
  __syncthreads();

  // Wave w = batch b: dot(mid[b,:], W2)
  int b = wave;
  float p = 0.0f;
  for (int j = lane; j < H_ / 2; j += 32) p += ml[b][j] * W2[j];
  for (int m = 16; m > 0; m >>= 1) p += __shfl_xor(p, m, 32);
  if (lane == 0) {
    float o = p + b2[0];
    out[b * T_ + t] = o;
    inp[b] = o;
  }
}

// ---------------------------------------------------------------------------
extern "C" void kernel_launch(void* const* d_in, const int* in_sizes, int n_in,
                              void* d_out, int out_size, void* d_ws, size_t ws_size,
                              hipStream_t stream) {
  const float* h0     = (const float*)d_in[0];
  const float* c0     = (const float*)d_in[1];
  const float* enc    = (const float*)d_in[2];
  const float* W_ih   = (const float*)d_in[3];
  const float* W_hh   = (const float*)d_in[4];
  const float* b_ih   = (const float*)d_in[5];
  const float* b_hh   = (const float*)d_in[6];
  const float* W_attn = (const float*)d_in[7];
  const float* b_attn = (const float*)d_in[8];
  const float* v      = (const float*)d_in[9];
  const float* W1     = (const float*)d_in[10];
  const float* b1     = (const float*)d_in[11];
  const float* W2     = (const float*)d_in[12];
  const float* b2     = (const float*)d_in[13];
  float* out = (float*)d_out;

  // Workspace layout (floats)
  float* ws = (float*)d_ws;
  float* enc_proj = ws;                    size_t off = (size_t)B_ * S_ * H_;
  float* hbuf0 = ws + off;  off += B_ * H_;
  float* hbuf1 = ws + off;  off += B_ * H_;
  float* c     = ws + off;  off += B_ * H_;
  float* q     = ws + off;  off += B_ * H_;
  float* ctx   = ws + off;  off += B_ * H_;
  float* xcat  = ws + off;  off += B_ * 2 * H_;
  float* inp   = ws + off;  off += 32;
  (void)ws_size; (void)in_sizes; (void)n_in; (void)out_size;

  // Fresh state every call
  k_init<<<(B_ * H_ + 255) / 256, 256, 0, stream>>>(h0, c0, hbuf0, c, inp);

  // One-time enc_proj GEMM: 8192 tiles, 4 waves/block
  k_enc_proj<<<2048, 128, 0, stream>>>(enc, W_attn, enc_proj);

  for (int t = 0; t < T_; ++t) {
    float* h_cur  = (t & 1) ? hbuf1 : hbuf0;
    float* h_next = (t & 1) ? hbuf0 : hbuf1;
    k_query<<<1, 1024, 0, stream>>>(h_cur, W_attn, b_attn, q);
    k_attn<<<B_, 512, 0, stream>>>(enc_proj, q, v, enc, ctx, xcat);
    k_gates_lstm<<<32, 128, 0, stream>>>(ctx, h_cur, inp, W_ih, W_hh, b_ih, b_hh,
                                         c, h_next, xcat);
    k_head<<<1, 512, 0, stream>>>(xcat, W1, b1, W2, b2, out, inp, t);
  }
}